// DGLJTNNEncoder_70849780515511
// MI455X (gfx1250) — compile-verified
//
#include <hip/hip_runtime.h>
#include <math.h>

typedef __attribute__((ext_vector_type(2))) float v2f;
typedef __attribute__((ext_vector_type(8))) float v8f;

#define HID    450
#define K2     900           // 2*HID, K-dim of all three fused GEMMs
#define KB     (K2 / 4)      // 225 K-blocks of 4
#define NP     512           // HID padded so col-tiles (32) divide evenly over 16 waves
#define NTILES 32            // NP/16
#define BLOCK  512           // 16 wave32 per block
#define TPW    2             // col-tiles per wave, compile-time uniform
#define TREES  256
#define NPT    16            // nodes per tree
#define MAXLVL 32            // non-backtracking walk on 16-node tree <= 15 edges

static __device__ __forceinline__ v8f zero8() {
    v8f v;
#pragma unroll
    for (int i = 0; i < 8; ++i) v[i] = 0.0f;
    return v;
}

static __device__ __forceinline__ float sigmoidf_(float x) {
    return 1.0f / (1.0f + __expf(-x));
}

// A fragment: 16x4 f32, lanes 0-15 hold K=k,k+1 ; lanes 16-31 hold K=k+2,k+3 (ISA 7.12.2)
// Two contiguous LDS floats per lane -> compiler emits one ds_load_2addr/b64.
static __device__ __forceinline__ v2f a_frag(const float* A, int lane, int k) {
    int r  = lane & 15;
    int kk = k + ((lane >> 4) << 1);
    v2f a;
    a.x = A[r * K2 + kk];
    a.y = A[r * K2 + kk + 1];
    return a;
}

// B fragment from fragment-major packed panel: one aligned 8-byte load per lane.
// packed[((kb*NP + n)*2 + p)*2 + {0,1}] = W[4*kb + 2*p + {0,1}][n],  p = lane>>4
static __device__ __forceinline__ v2f b_fragp(const float* __restrict__ Wp, int lane,
                                              int kb, int n0) {
    int n = n0 + (lane & 15);
    int p = lane >> 4;
    size_t idx = ((((size_t)kb * NP + n) << 1) + p) << 1;
    const float2* src = (const float2*)(Wp + idx);
    float2 d = *src;
    v2f b;
    b.x = d.x;
    b.y = d.y;
    return b;
}

static __device__ __forceinline__ v8f wmma_f32(v2f a, v2f b, v8f c) {
    return __builtin_amdgcn_wmma_f32_16x16x4_f32(false, a, false, b, (short)0, c, false, false);
}

// Straight-line L2 warm for one weight panel (no branches around WMMA loops)
static __device__ __forceinline__ void warm_panel(const float* __restrict__ W, int tid) {
#pragma unroll 1
    for (size_t ofs = (size_t)tid * 16; ofs < (size_t)K2 * NP; ofs += (size_t)BLOCK * 16)
        __builtin_prefetch(W + ofs, 0, 1);
}

// ---------------------------------------------------------------- utility kernels

__global__ void zero_kernel(float* __restrict__ p, size_t n) {
    size_t i = (size_t)blockIdx.x * blockDim.x + threadIdx.x;
    size_t stride = (size_t)gridDim.x * blockDim.x;
    for (; i < n; i += stride) p[i] = 0.0f;
}

__global__ void embed_kernel(const int* __restrict__ wid, const float* __restrict__ emb,
                             float* __restrict__ x) {
    int v = blockIdx.x;
    const float* e = emb + (size_t)wid[v] * HID;
    float* xr = x + (size_t)v * HID;
    for (int c = threadIdx.x; c < HID; c += blockDim.x) xr[c] = e[c];
}

// Pack weights into fragment-major swizzled panels (zero-padded cols HID..NP),
// Wur = vstack(Wr, Ur); pad biases to NP.
__global__ void pack_kernel(const float* __restrict__ Wz,
                            const float* __restrict__ bz,
                            const float* __restrict__ Wr, const float* __restrict__ Ur,
                            const float* __restrict__ bur,
                            const float* __restrict__ Wh, const float* __restrict__ bh,
                            const float* __restrict__ Wg, const float* __restrict__ bg,
                            float* __restrict__ Wz_p, float* __restrict__ bz_p,
                            float* __restrict__ Wh_p, float* __restrict__ bh_p,
                            float* __restrict__ Wur_p, float* __restrict__ bur_p,
                            float* __restrict__ Wg_p, float* __restrict__ bg_p) {
    // one thread per lane-pair t = (kb*NP + n)*2 + p  ->  writes floats 2t, 2t+1
    size_t pairs = (size_t)KB * NP * 2;
    size_t i = (size_t)blockIdx.x * blockDim.x + threadIdx.x;
    size_t stride = (size_t)gridDim.x * blockDim.x;
    for (size_t t = i; t < pairs; t += stride) {
        int p = (int)(t & 1);
        size_t u = t >> 1;
        int n  = (int)(u % NP);
        int kb = (int)(u / NP);
        int k0 = 4 * kb + 2 * p;          // source rows k0, k0+1
        bool ok = (n < HID);
        float z0 = 0.f, z1 = 0.f, h0 = 0.f, h1 = 0.f, g0 = 0.f, g1 = 0.f, u0 = 0.f, u1 = 0.f;
        if (ok) {
            z0 = Wz[(size_t)k0 * HID + n];       z1 = Wz[(size_t)(k0 + 1) * HID + n];
            h0 = Wh[(size_t)k0 * HID + n];       h1 = Wh[(size_t)(k0 + 1) * HID + n];
            g0 = Wg[(size_t)k0 * HID + n];       g1 = Wg[(size_t)(k0 + 1) * HID + n];
            u0 = (k0 < HID)     ? Wr[(size_t)k0 * HID + n]       : Ur[(size_t)(k0 - HID) * HID + n];
            u1 = (k0 + 1 < HID) ? Wr[(size_t)(k0 + 1) * HID + n] : Ur[(size_t)(k0 + 1 - HID) * HID + n];
        }
        Wz_p[2 * t] = z0;  Wz_p[2 * t + 1] = z1;
        Wh_p[2 * t] = h0;  Wh_p[2 * t + 1] = h1;
        Wg_p[2 * t] = g0;  Wg_p[2 * t + 1] = g1;
        Wur_p[2 * t] = u0; Wur_p[2 * t + 1] = u1;
    }
    if (i < NP) {
        int n = (int)i;
        bool ok = (n < HID);
        bz_p[n]  = ok ? bz[n]  : 0.0f;
        bh_p[n]  = ok ? bh[n]  : 0.0f;
        bur_p[n] = ok ? bur[n] : 0.0f;
        bg_p[n]  = ok ? bg[n]  : 0.0f;
    }
}

// ---------------------------------------------------------------- per-level kernel
// grid.x = E1 (local edge), grid.y = TREES/16 (tree tile); BLOCK threads = 16 wave32.
// LDS: Az[16][K2] = [src_x | s], Ah[16][K2] = [src_x | accum_rm] (reused as [dst_x | m_new]).
__global__ void __launch_bounds__(BLOCK)
level_kernel(const float* __restrict__ x, float* __restrict__ m, float* __restrict__ rm,
             const int* __restrict__ edge_src, const int* __restrict__ edge_dst,
             const int* __restrict__ lg_src, const int* __restrict__ lg_dst,
             const int* __restrict__ edge_level,
             const float* __restrict__ Wz_p, const float* __restrict__ bz_p,
             const float* __restrict__ Wh_p, const float* __restrict__ bh_p,
             const float* __restrict__ Wur_p, const float* __restrict__ bur_p,
             int lvl, int E1, int L1) {
    int j = blockIdx.x;                 // local edge within tree pattern (same for all trees)
    if (edge_level[j] != lvl) return;   // block-uniform exit: EXEC all-ones around WMMA
    int rtile = blockIdx.y;

    extern __shared__ float lds[];
    float* Az = lds;                    // 16 x K2
    float* Ah = lds + 16 * K2;          // 16 x K2, later reused as Ar

    int tid  = threadIdx.x;
    int row  = tid >> 5;                // 0..15 (tree within tile)
    int cl   = tid & 31;                // column lane for staging
    int tree = rtile * 16 + row;
    int e    = tree * E1 + j;

    // ---- stage A matrices: segment sums in fixed order (deterministic)
    {
        const float* sx = x + (size_t)edge_src[e] * HID;
        for (int c = cl; c < HID; c += 32) {
            float v = sx[c];
            Az[row * K2 + c] = v;
            Ah[row * K2 + c] = v;
            Az[row * K2 + HID + c] = 0.0f;
            Ah[row * K2 + HID + c] = 0.0f;
        }
        int base = tree * L1;
        for (int i = 0; i < L1; ++i) {
            if (lg_dst[base + i] == e) {
                const float* mp  = m  + (size_t)lg_src[base + i] * HID;
                const float* rmp = rm + (size_t)lg_src[base + i] * HID;
                for (int c = cl; c < HID; c += 32) {
                    Az[row * K2 + HID + c] += mp[c];
                    Ah[row * K2 + HID + c] += rmp[c];
                }
            }
        }
        warm_panel(Wz_p, tid);          // L2 warm, straight-line, before the barrier
        warm_panel(Wh_p, tid);
    }
    __syncthreads();

    int lane = tid & 31;
    int w    = tid >> 5;                // wave id; owns col-tiles 2w, 2w+1 (uniform, no branch)
    int n0 = (2 * w) * 16, n1 = (2 * w + 1) * 16;

    v8f zacc[TPW], hacc[TPW];
#pragma unroll
    for (int i = 0; i < TPW; ++i) { zacc[i] = zero8(); hacc[i] = zero8(); }

    // ---- GEMM1 (z-gate) + GEMM2 (candidate); single-BB loop body
    for (int kb = 0; kb < KB; ++kb) {
        v2f az = a_frag(Az, lane, 4 * kb);
        v2f ah = a_frag(Ah, lane, 4 * kb);
        zacc[0] = wmma_f32(az, b_fragp(Wz_p, lane, kb, n0), zacc[0]);
        zacc[1] = wmma_f32(az, b_fragp(Wz_p, lane, kb, n1), zacc[1]);
        hacc[0] = wmma_f32(ah, b_fragp(Wh_p, lane, kb, n0), hacc[0]);
        hacc[1] = wmma_f32(ah, b_fragp(Wh_p, lane, kb, n1), hacc[1]);
    }
    __syncthreads();                     // GEMM reads of Ah done; safe to overwrite

    // ---- stage dst_x into Ar[:, 0:HID)
    {
        const float* dx = x + (size_t)edge_dst[e] * HID;
        for (int c = cl; c < HID; c += 32) Ah[row * K2 + c] = dx[c];
    }

    // ---- elementwise: z = sigma(.), pre_m = tanh(.), m_new = (1-z)*s + z*pre_m
    float mnew[TPW][8];
#pragma unroll
    for (int i = 0; i < TPW; ++i) {
        int col = (2 * w + i) * 16 + (lane & 15);
#pragma unroll
        for (int r = 0; r < 8; ++r) {
            mnew[i][r] = 0.0f;
            int rr = r + ((lane >> 4) << 3);           // output row 0..15
            if (col < HID) {
                float zv = sigmoidf_(zacc[i][r] + bz_p[col]);
                float pm = tanhf(hacc[i][r] + bh_p[col]);
                float sv = Az[rr * K2 + HID + col];
                float mn = (1.0f - zv) * sv + zv * pm;
                mnew[i][r] = mn;
                Ah[rr * K2 + HID + col] = mn;          // Ar[:, HID:2*HID) = m_new
                int erow = (rtile * 16 + rr) * E1 + j;
                m[(size_t)erow * HID + col] = mn;
            }
        }
    }
    __syncthreads();

    // ---- GEMM3: r_pre = [dst_x | m_new] @ [Wr;Ur]
    v8f racc[TPW];
#pragma unroll
    for (int i = 0; i < TPW; ++i) racc[i] = zero8();
    for (int kb = 0; kb < KB; ++kb) {
        v2f ar = a_frag(Ah, lane, 4 * kb);
        racc[0] = wmma_f32(ar, b_fragp(Wur_p, lane, kb, n0), racc[0]);
        racc[1] = wmma_f32(ar, b_fragp(Wur_p, lane, kb, n1), racc[1]);
    }

    // ---- rm = sigma(r_pre + bur) * m_new
#pragma unroll
    for (int i = 0; i < TPW; ++i) {
        int col = (2 * w + i) * 16 + (lane & 15);
        if (col < HID) {
#pragma unroll
            for (int r = 0; r < 8; ++r) {
                int rr = r + ((lane >> 4) << 3);
                float rv = sigmoidf_(racc[i][r] + bur_p[col]);
                int erow = (rtile * 16 + rr) * E1 + j;
                rm[(size_t)erow * HID + col] = rv * mnew[i][r];
            }
        }
    }
}

// ---------------------------------------------------------------- readout kernel
// One block per tree (16 nodes): h = relu([x | m_node] @ Wg + bg); also scatter roots.
__global__ void __launch_bounds__(BLOCK)
out_kernel(const float* __restrict__ x, const float* __restrict__ m,
           const int* __restrict__ edge_dst, const int* __restrict__ root_ids,
           const float* __restrict__ Wg_p, const float* __restrict__ bg_p,
           float* __restrict__ out, int E1, int n_nodes) {
    int rtile = blockIdx.x;             // == tree index (NPT == 16 rows per tile)
    extern __shared__ float lds[];
    float* Ag = lds;                    // 16 x K2

    int tid  = threadIdx.x;
    int row  = tid >> 5;
    int cl   = tid & 31;
    int node = rtile * NPT + row;

    {
        const float* xp = x + (size_t)node * HID;
        for (int c = cl; c < HID; c += 32) {
            Ag[row * K2 + c] = xp[c];
            Ag[row * K2 + HID + c] = 0.0f;
        }
        int base = rtile * E1;          // edges of this tree
        for (int jj = 0; jj < E1; ++jj) {
            if (edge_dst[base + jj] == node) {
                const float* mp = m + (size_t)(base + jj) * HID;
                for (int c = cl; c < HID; c += 32) Ag[row * K2 + HID + c] += mp[c];
            }
        }
        warm_panel(Wg_p, tid);
    }
    __syncthreads();

    int lane = tid & 31;
    int w    = tid >> 5;
    int n0 = (2 * w) * 16, n1 = (2 * w + 1) * 16;
    v8f gacc[TPW];
#pragma unroll
    for (int i = 0; i < TPW; ++i) gacc[i] = zero8();

    for (int kb = 0; kb < KB; ++kb) {
        v2f ag = a_frag(Ag, lane, 4 * kb);
        gacc[0] = wmma_f32(ag, b_fragp(Wg_p, lane, kb, n0), gacc[0]);
        gacc[1] = wmma_f32(ag, b_fragp(Wg_p, lane, kb, n1), gacc[1]);
    }

#pragma unroll
    for (int i = 0; i < TPW; ++i) {
        int col = (2 * w + i) * 16 + (lane & 15);
        if (col < HID) {
#pragma unroll
            for (int r = 0; r < 8; ++r) {
                int rr = r + ((lane >> 4) << 3);
                int nr = rtile * NPT + rr;
                float h = fmaxf(gacc[i][r] + bg_p[col], 0.0f);
                out[(size_t)nr * HID + col] = h;
                if (root_ids[rtile] == nr)
                    out[(size_t)n_nodes * HID + (size_t)rtile * HID + col] = h;
            }
        }
    }
}

// ---------------------------------------------------------------- launch

extern "C" void kernel_launch(void* const* d_in, const int* in_sizes, int n_in,
                              void* d_out, int out_size, void* d_ws, size_t ws_size,
                              hipStream_t stream) {
    const int*   wid        = (const int*)d_in[0];
    const int*   edge_src   = (const int*)d_in[1];
    const int*   edge_dst   = (const int*)d_in[2];
    const int*   lg_src     = (const int*)d_in[3];
    const int*   lg_dst     = (const int*)d_in[4];
    const int*   edge_level = (const int*)d_in[5];
    const int*   root_ids   = (const int*)d_in[6];
    /* d_in[7] = num_levels (device scalar; fixed MAXLVL chain used instead) */
    const float* emb = (const float*)d_in[8];
    const float* Wz  = (const float*)d_in[9];
    const float* bz  = (const float*)d_in[10];
    const float* Wr  = (const float*)d_in[11];
    const float* Ur  = (const float*)d_in[12];
    const float* bur = (const float*)d_in[13];
    const float* Wh  = (const float*)d_in[14];
    const float* bh  = (const float*)d_in[15];
    const float* Wg  = (const float*)d_in[16];
    const float* bg  = (const float*)d_in[17];

    int n_nodes = in_sizes[0];
    int n_edges = in_sizes[1];
    int n_lg    = in_sizes[3];
    int E1      = n_edges / TREES;
    int L1      = n_lg / TREES;

    float* ws = (float*)d_ws;
    size_t off = 0;
    float* x     = ws + off; off += (size_t)n_nodes * HID;
    float* m     = ws + off; off += (size_t)n_edges * HID;
    float* rm    = ws + off; off += (size_t)n_edges * HID;
    float* Wz_p  = ws + off; off += (size_t)K2 * NP;
    float* Wh_p  = ws + off; off += (size_t)K2 * NP;
    float* Wur_p = ws + off; off += (size_t)K2 * NP;
    float* Wg_p  = ws + off; off += (size_t)K2 * NP;
    float* bz_p  = ws + off; off += NP;
    float* bh_p  = ws + off; off += NP;
    float* bur_p = ws + off; off += NP;
    float* bg_p  = ws + off; off += NP;
    (void)ws_size; (void)n_in; (void)out_size;

    // m and rm are contiguous -> single zero pass (must re-init every call)
    zero_kernel<<<2048, 256, 0, stream>>>(m, (size_t)2 * n_edges * HID);
    embed_kernel<<<n_nodes, 256, 0, stream>>>(wid, emb, x);
    pack_kernel<<<2048, 256, 0, stream>>>(Wz, bz, Wr, Ur, bur, Wh, bh, Wg, bg,
                                          Wz_p, bz_p, Wh_p, bh_p, Wur_p, bur_p, Wg_p, bg_p);

    size_t lds2 = (size_t)2 * 16 * K2 * sizeof(float);   // 115200 B (< 320KB/WGP on CDNA5)
    size_t lds1 = (size_t)16 * K2 * sizeof(float);       // 57600 B
    dim3 lvlGrid(E1, TREES / 16);
    for (int lvl = 0; lvl < MAXLVL; ++lvl) {
        level_kernel<<<lvlGrid, BLOCK, lds2, stream>>>(
            x, m, rm, edge_src, edge_dst, lg_src, lg_dst, edge_level,
            Wz_p, bz_p, Wh_p, bh_p, Wur_p, bur_p, lvl, E1, L1);
    }
    out_kernel<<<n_nodes / NPT, BLOCK, lds1, stream>>>(
        x, m, edge_dst, root_ids, Wg_p, bg_p, (float*)d_out, E1, n_nodes);
}